// ACBLoss3D_15040975470950
// MI455X (gfx1250) — compile-verified
//
#include <hip/hip_runtime.h>

// ---------------- problem constants ----------------
#define TSTEPS   1000
#define IMG_B    32
#define IMG_H    512
#define IMG_W    512
#define IMG_HW   (IMG_H * IMG_W)                              // 262144
#define N_IMG    ((unsigned long long)IMG_B * IMG_HW)         // 8,388,608
#define N_TR     ((unsigned long long)IMG_B * IMG_H * TSTEPS) // 16,384,000

typedef unsigned int       u32;
typedef unsigned long long u64;
typedef u32 u32x4 __attribute__((ext_vector_type(4)));
typedef u32 u32x8 __attribute__((ext_vector_type(8)));

// Workgroup-relative LDS byte offset of a __shared__ object.
// The LDS aperture spans a full 4GB region,
// so the low 32 bits of the generic address are exactly the lds_addr the
// async/TDM instructions want in their VDST / D#.lds_addr fields.
__device__ __forceinline__ u32 lds_addr_of(const void* p) {
    return (u32)(u64)p;
}

// q = int(v*1000) - 1 ; negative wraps (jnp q % timesteps for nonzero pixels)
__device__ __forceinline__ int holo_bin(float v) {
    int q = (int)(v * 1000.0f) - 1;
    return (q < 0) ? q + TSTEPS : q;
}

// ---------------------------------------------------------------------------
// TDM descriptor helpers (CDNA5 ISA): 1-D tile of `nelem` 4-byte words.
// Group0: count=1 | lds_addr | global_addr[56:0] | type=2 ("image")
// Group1: data_size=4B, tensor_dim0 = tile_dim0 = nelem, dim0 stride = nelem.
// ---------------------------------------------------------------------------
__device__ __forceinline__ u32x4 tdm_group0(u32 lds_byte_addr, u64 gaddr) {
    u32x4 g;
    g.x = 0x1u;                                    // count=1, user desc, no gather
    g.y = lds_byte_addr;                           // lds_addr [63:32]
    g.z = (u32)(gaddr & 0xFFFFFFFFull);            // global_addr [95:64]
    g.w = (u32)((gaddr >> 32) & 0x01FFFFFFull)     // global_addr [120:96]
        | (2u << 30);                              // type=2  [127:126]
    return g;
}

__device__ __forceinline__ u32x8 tdm_group1_1d(u32 nelem) {
    u32x8 g;
    g.s0 = (2u << 16);                 // wg_mask=0, data_size=2 (4B), no flags
    g.s1 = (nelem & 0xFFFFu) << 16;    // tensor_dim0[15:0] (barrier addr = 0)
    g.s2 = ((nelem >> 16) & 0xFFFFu)   // tensor_dim0[31:16]
         | (1u << 16);                 // tensor_dim1 = 1
    g.s3 = (nelem & 0xFFFFu) << 16;    // tile_dim0 = nelem
    g.s4 = 0u;                         // tile_dim1 = 0 (unused), tile_dim2 = 0
    g.s5 = nelem;                      // tensor_dim0_stride[31:0]
    g.s6 = 0u;                         // stride hi, tensor_dim1_stride lo
    g.s7 = 0u;
    return g;
}

// ---------------------------------------------------------------------------
// Kernel 1: direct ACB-MSE partials over the raw images (coalesced float4
// loads + global_prefetch_b8; also warms the 192MB L2 so both 33.5MB images
// stay cache-resident for the two transform passes).
// ---------------------------------------------------------------------------
__global__ void acb_direct_kernel(const float4* __restrict__ r,
                                  const float4* __restrict__ t,
                                  double* __restrict__ sums,
                                  unsigned int* __restrict__ cnt1,
                                  int n4)
{
    __shared__ float red0[256];
    __shared__ float red1[256];
    __shared__ int   redc[256];

    const int tid    = threadIdx.x;
    const int stride = gridDim.x * blockDim.x;

    float a0 = 0.0f, a1 = 0.0f;
    int   c1 = 0;

    for (int i = blockIdx.x * blockDim.x + tid; i < n4; i += stride) {
        if (i + stride < n4) {
            __builtin_prefetch(&r[i + stride], 0, 0);
            __builtin_prefetch(&t[i + stride], 0, 0);
        }
        float4 rv = r[i];
        float4 tv = t[i];
        { float d = rv.x - tv.x, d2 = d * d;
          if (tv.x == 0.0f) a0 += d2; else { a1 += d2; c1++; } }
        { float d = rv.y - tv.y, d2 = d * d;
          if (tv.y == 0.0f) a0 += d2; else { a1 += d2; c1++; } }
        { float d = rv.z - tv.z, d2 = d * d;
          if (tv.z == 0.0f) a0 += d2; else { a1 += d2; c1++; } }
        { float d = rv.w - tv.w, d2 = d * d;
          if (tv.w == 0.0f) a0 += d2; else { a1 += d2; c1++; } }
    }

    red0[tid] = a0; red1[tid] = a1; redc[tid] = c1;
    __syncthreads();
    for (int s = blockDim.x >> 1; s > 0; s >>= 1) {
        if (tid < s) {
            red0[tid] += red0[tid + s];
            red1[tid] += red1[tid + s];
            redc[tid] += redc[tid + s];
        }
        __syncthreads();
    }
    if (tid == 0) {
        atomicAdd(&sums[0], (double)red0[0]);
        atomicAdd(&sums[1], (double)red1[0]);
        atomicAdd(cnt1, (unsigned int)redc[0]);
    }
}

// ---------------------------------------------------------------------------
// Shared tail: scatter from LDS stage + class-balanced MSE over 1000 bins.
// ---------------------------------------------------------------------------
__device__ __forceinline__ void holo_scatter_and_mse(
    const float* stage_r, const float* stage_t,
    int* rbins, int* tbins,
    float* red0, float* red1, int* redc,
    double* sums, unsigned int* cnt1, int nscan)
{
    const int tid = threadIdx.x;

    for (int i = tid; i < nscan; i += blockDim.x) {
        float rv = stage_r[i];
        float tv = stage_t[i];
        if (rv != 0.0f) atomicMax(&rbins[holo_bin(rv)], i);  // ds_max_i32
        if (tv != 0.0f) atomicMax(&tbins[holo_bin(tv)], i);
    }
    __syncthreads();

    float a0 = 0.0f, a1 = 0.0f;
    int   c1 = 0;
    for (int k = tid; k < TSTEPS; k += blockDim.x) {
        float rv = (float)rbins[k];
        float tv = (float)tbins[k];
        float d  = rv - tv;
        float d2 = d * d;
        if (tbins[k] == 0) a0 += d2; else { a1 += d2; c1++; }
    }

    red0[tid] = a0; red1[tid] = a1; redc[tid] = c1;
    __syncthreads();
    for (int s = blockDim.x >> 1; s > 0; s >>= 1) {
        if (tid < s) {
            red0[tid] += red0[tid + s];
            red1[tid] += red1[tid + s];
            redc[tid] += redc[tid + s];
        }
        __syncthreads();
    }
    if (tid == 0) {
        atomicAdd(&sums[0], (double)red0[0]);
        atomicAdd(&sums[1], (double)red1[0]);
        atomicAdd(cnt1, (unsigned int)redc[0]);
    }
}

// ---------------------------------------------------------------------------
// Kernel 2a (vx, transpose=True): line j = original column, scan rows i.
// Per-lane 2KB-strided column gather goes through GLOBAL_LOAD_ASYNC_TO_LDS_B32
// (ASYNCcnt path): issue all async loads, zero the bins while they fly,
// s_wait_asynccnt, barrier, then scatter-max from the LDS stage.
// ---------------------------------------------------------------------------
__global__ void holo_mse_x_kernel(const float* __restrict__ recon,
                                  const float* __restrict__ target,
                                  double* __restrict__ sums,
                                  unsigned int* __restrict__ cnt1)
{
    __shared__ float stage_r[IMG_H];
    __shared__ float stage_t[IMG_H];
    __shared__ int   rbins[TSTEPS];
    __shared__ int   tbins[TSTEPS];
    __shared__ float red0[256];
    __shared__ float red1[256];
    __shared__ int   redc[256];

    const int tid = threadIdx.x;
    const int b   = blockIdx.x >> 9;   // / 512
    const int j   = blockIdx.x & 511;  // % 512  (original column)

    const u64 rbase = (u64)(recon  + (size_t)b * IMG_HW);
    const u64 tbase = (u64)(target + (size_t)b * IMG_HW);
    const u32 ldsr  = lds_addr_of(stage_r);
    const u32 ldst  = lds_addr_of(stage_t);

    // async gather of the column into LDS (element i lives at i*W + j)
    for (int i = tid; i < IMG_H; i += blockDim.x) {
        u32 boff = (u32)((i * IMG_W + j) * 4);
        asm volatile("global_load_async_to_lds_b32 %0, %1, %2"
                     :: "v"(ldsr + (u32)i * 4u), "v"(boff), "s"(rbase)
                     : "memory");
        asm volatile("global_load_async_to_lds_b32 %0, %1, %2"
                     :: "v"(ldst + (u32)i * 4u), "v"(boff), "s"(tbase)
                     : "memory");
    }

    // zero the time bins while the async loads are in flight
    for (int k = tid; k < TSTEPS; k += blockDim.x) {
        rbins[k] = 0;
        tbins[k] = 0;
    }

    asm volatile("s_wait_asynccnt 0x0" ::: "memory");
    __syncthreads();

    holo_scatter_and_mse(stage_r, stage_t, rbins, tbins,
                         red0, red1, redc, sums, cnt1, IMG_H);
}

// ---------------------------------------------------------------------------
// Kernel 2b (vy, transpose=False): line j = original row, scan columns i.
// The row is one contiguous 2KB block -> Tensor Data Mover. Wave 0 issues two
// 1-D TDM descriptors (recon row, target row), bins are zeroed while the DMA
// flies, then s_wait_tensorcnt + barrier.
// ---------------------------------------------------------------------------
__global__ void holo_mse_y_kernel(const float* __restrict__ recon,
                                  const float* __restrict__ target,
                                  double* __restrict__ sums,
                                  unsigned int* __restrict__ cnt1)
{
    __shared__ float stage_r[IMG_W];
    __shared__ float stage_t[IMG_W];
    __shared__ int   rbins[TSTEPS];
    __shared__ int   tbins[TSTEPS];
    __shared__ float red0[256];
    __shared__ float red1[256];
    __shared__ int   redc[256];

    const int tid = threadIdx.x;
    const int b   = blockIdx.x >> 9;   // / 512
    const int j   = blockIdx.x & 511;  // % 512  (original row)

    const float* rrow = recon  + (size_t)b * IMG_HW + (size_t)j * IMG_W;
    const float* trow = target + (size_t)b * IMG_HW + (size_t)j * IMG_W;

    if (tid < 32) {  // one wave issues the DMAs (tensor ops are per-wave)
        u32x4 g0r = tdm_group0(lds_addr_of(stage_r), (u64)rrow);
        u32x4 g0t = tdm_group0(lds_addr_of(stage_t), (u64)trow);
        u32x8 g1  = tdm_group1_1d(IMG_W);
        asm volatile("tensor_load_to_lds %0, %1" :: "s"(g0r), "s"(g1) : "memory");
        asm volatile("tensor_load_to_lds %0, %1" :: "s"(g0t), "s"(g1) : "memory");
    }

    // zero the time bins while the TDM transfers are in flight
    for (int k = tid; k < TSTEPS; k += blockDim.x) {
        rbins[k] = 0;
        tbins[k] = 0;
    }

    __builtin_amdgcn_s_wait_tensorcnt(0);
    __syncthreads();

    holo_scatter_and_mse(stage_r, stage_t, rbins, tbins,
                         red0, red1, redc, sums, cnt1, IMG_W);
}

// ---------------------------------------------------------------------------
// Kernel 3: combine the three terms exactly like the reference
// (each class mean contributes 0 if its count is 0; all weights are 1.0).
// ---------------------------------------------------------------------------
__global__ void acb_finalize_kernel(const double* __restrict__ sums,      // 6
                                    const unsigned int* __restrict__ cnts, // 3
                                    float* __restrict__ out)
{
    auto term = [](double s0, double s1, unsigned long long c1v,
                   unsigned long long n) -> double {
        unsigned long long c0v = n - c1v;
        double l0 = (c0v > 0) ? (s0 / (double)c0v) : 0.0;
        double l1 = (c1v > 0) ? (s1 / (double)c1v) : 0.0;
        return l0 + l1;
    };
    double vt = term(sums[0], sums[1], cnts[0], N_IMG);
    double vx = term(sums[2], sums[3], cnts[1], N_TR);
    double vy = term(sums[4], sums[5], cnts[2], N_TR);
    out[0] = (float)(vt + vx + vy);
}

// ---------------------------------------------------------------------------
extern "C" void kernel_launch(void* const* d_in, const int* in_sizes, int n_in,
                              void* d_out, int out_size, void* d_ws, size_t ws_size,
                              hipStream_t stream)
{
    const float* recon  = (const float*)d_in[0];  // reconstructed_image
    const float* target = (const float*)d_in[1];  // target_image
    float* out = (float*)d_out;

    // ws layout: 6 doubles (48 B) then 3 uints (12 B)
    double*       sums = (double*)d_ws;
    unsigned int* cnts = (unsigned int*)((char*)d_ws + 48);

    hipMemsetAsync(d_ws, 0, 64, stream); // graph-capturable

    const int n4 = (int)(N_IMG / 4);
    acb_direct_kernel<<<2048, 256, 0, stream>>>(
        (const float4*)recon, (const float4*)target, sums + 0, cnts + 0, n4);

    // vx: transpose=True (async-to-LDS column gather)
    holo_mse_x_kernel<<<IMG_B * IMG_W, 256, 0, stream>>>(
        recon, target, sums + 2, cnts + 1);

    // vy: transpose=False (TDM row DMA)
    holo_mse_y_kernel<<<IMG_B * IMG_H, 256, 0, stream>>>(
        recon, target, sums + 4, cnts + 2);

    acb_finalize_kernel<<<1, 1, 0, stream>>>(sums, cnts, out);
}